// Net_7610682049228
// MI455X (gfx1250) — compile-verified
//
#include <hip/hip_runtime.h>

typedef __attribute__((ext_vector_type(2))) float v2f;
typedef __attribute__((ext_vector_type(8))) float v8f;

#define N_PTS   4096
#define JCHUNK  1024
#define THREADS 256
#define WAVES   8                 // 256 threads / wave32
#define I_PER_BLOCK (WAVES * 16)  // 128
#define NBLOCKS (N_PTS / I_PER_BLOCK)  // 32

// exp(-0.5*t^2) == exp2(C2 * t^2), C2 = -0.5*log2(e)
#define C2_CONST (-0.72134752044448170368f)

// One fused kernel:
//  - per-lane: project its query row x[i] -> Zw/h (zs[3])
//  - cooperative: project train_X chunk -> Xw/h and stage [Xs|Y] in LDS
//  - per wave: 16 i-rows; j-tiles of 4 accumulated with V_WMMA_F32_16X16X4_F32
//      A[16x4]  = K[i, j0..j0+3, d]   (exp2 computed in VALU)
//      B[4x16]  = col0: Y[j,d], col1: 1.0, cols 2..15: 0  (built via fma, branchless)
//      C[16x16] accumulates: col0 = numerator, col1 = denominator
//  - leave-one-out handled OUTSIDE the hot loop: the diagonal term K(i,i,d)
//    is recomputed exactly (identical FP expression as staging) and subtracted
//    from num/den before the final divide.
__global__ __launch_bounds__(THREADS) void nw_loo_kernel(
    const float* __restrict__ x,      // [4096,4]
    const float* __restrict__ trainX, // [4096,4]
    const float* __restrict__ Y,      // [4096,3]
    const float* __restrict__ W1,     // [3,4]
    const float* __restrict__ W2,     // [3,3]
    const float* __restrict__ hptr,   // [1]
    float* __restrict__ out)          // [4096,3]
{
    __shared__ float4 ldsXY[JCHUNK * 2];   // per j: [2j] = Xw/h (xyz), [2j+1] = Y (xyz)
    __shared__ float  sred[WAVES * 96];    // per-wave num/den spill: [d][0:16]=num,[16:32]=den

    const int tid  = threadIdx.x;
    const int lane = tid & 31;
    const int w    = tid >> 5;
    const int r    = lane & 15;   // i-row within wave tile == B column index n
    const int half = lane >> 4;   // selects K pair {0,1} vs {2,3}
    const int ibase = blockIdx.x * I_PER_BLOCK + w * 16;
    const int i     = ibase + r;

    // uniform weights -> SGPRs
    float w1[12], w2[9];
#pragma unroll
    for (int k = 0; k < 12; ++k) w1[k] = W1[k];
#pragma unroll
    for (int k = 0; k < 9; ++k)  w2[k] = W2[k];
    const float invh = 1.0f / hptr[0];

    // per-lane query projection: zs[d] = fc2(relu(fc1(x_i)))[d] / h
    float zs0, zs1, zs2;
    {
        const float4 xi = reinterpret_cast<const float4*>(x)[i];
        float h0 = fmaxf(w1[0]*xi.x + w1[1]*xi.y + w1[2] *xi.z + w1[3] *xi.w, 0.0f);
        float h1 = fmaxf(w1[4]*xi.x + w1[5]*xi.y + w1[6] *xi.z + w1[7] *xi.w, 0.0f);
        float h2 = fmaxf(w1[8]*xi.x + w1[9]*xi.y + w1[10]*xi.z + w1[11]*xi.w, 0.0f);
        zs0 = (w2[0]*h0 + w2[1]*h1 + w2[2]*h2) * invh;
        zs1 = (w2[3]*h0 + w2[4]*h1 + w2[5]*h2) * invh;
        zs2 = (w2[6]*h0 + w2[7]*h1 + w2[8]*h2) * invh;
    }

    // branchless B-column constants: b = selY*Y + addC
    const float selY = (r == 0) ? 1.0f : 0.0f;   // numerator column
    const float addC = (r == 1) ? 1.0f : 0.0f;   // denominator column

    v8f acc0 = {}, acc1 = {}, acc2 = {};

    for (int c = 0; c < N_PTS; c += JCHUNK) {
        // stage chunk: project train_X -> Xw/h, copy Y
        for (int j = tid; j < JCHUNK; j += THREADS) {
            const int gj = c + j;
            const float4 tx = reinterpret_cast<const float4*>(trainX)[gj];
            float h0 = fmaxf(w1[0]*tx.x + w1[1]*tx.y + w1[2] *tx.z + w1[3] *tx.w, 0.0f);
            float h1 = fmaxf(w1[4]*tx.x + w1[5]*tx.y + w1[6] *tx.z + w1[7] *tx.w, 0.0f);
            float h2 = fmaxf(w1[8]*tx.x + w1[9]*tx.y + w1[10]*tx.z + w1[11]*tx.w, 0.0f);
            ldsXY[j*2 + 0] = make_float4((w2[0]*h0 + w2[1]*h1 + w2[2]*h2) * invh,
                                         (w2[3]*h0 + w2[4]*h1 + w2[5]*h2) * invh,
                                         (w2[6]*h0 + w2[7]*h1 + w2[8]*h2) * invh, 0.0f);
            ldsXY[j*2 + 1] = make_float4(Y[gj*3 + 0], Y[gj*3 + 1], Y[gj*3 + 2], 0.0f);
        }
        __syncthreads();

        for (int t = 0; t < JCHUNK; t += 4) {
            const int la = t + 2 * half;   // this lane's first j within chunk
            const float4 xa = ldsXY[la*2 + 0];
            const float4 ya = ldsXY[la*2 + 1];
            const float4 xb = ldsXY[la*2 + 2];
            const float4 yb = ldsXY[la*2 + 3];

#define DO_D(ACC, XA, XB, YA, YB, ZS)                                          \
            {                                                                  \
                float ta = (XA) - (ZS);                                        \
                float aA = __builtin_amdgcn_exp2f(C2_CONST * ta * ta);         \
                float tb = (XB) - (ZS);                                        \
                float aB = __builtin_amdgcn_exp2f(C2_CONST * tb * tb);         \
                v2f Am = {aA, aB};                                             \
                v2f Bm = {fmaf(selY, (YA), addC), fmaf(selY, (YB), addC)};     \
                ACC = __builtin_amdgcn_wmma_f32_16x16x4_f32(                   \
                        false, Am, false, Bm, (short)0, ACC, false, false);    \
            }

            DO_D(acc0, xa.x, xb.x, ya.x, yb.x, zs0)
            DO_D(acc1, xa.y, xb.y, ya.y, yb.y, zs1)
            DO_D(acc2, xa.z, xb.z, ya.z, yb.z, zs2)
#undef DO_D
        }
        __syncthreads();
    }

    // Extract num (col 0) / den (col 1) from C layout:
    // VGPR v, lanes 0-15: D[M=v][N=lane]; lanes 16-31: D[M=v+8][N=lane-16]
    float* sw = &sred[w * 96];
    {
        v8f accs[3] = {acc0, acc1, acc2};
#pragma unroll
        for (int d = 0; d < 3; ++d) {
            int base = -1;
            if      (lane == 0)  base = d*32 + 0;    // num, M=0..7
            else if (lane == 1)  base = d*32 + 16;   // den, M=0..7
            else if (lane == 16) base = d*32 + 8;    // num, M=8..15
            else if (lane == 17) base = d*32 + 24;   // den, M=8..15
            if (base >= 0) {
#pragma unroll
                for (int v = 0; v < 8; ++v) sw[base + v] = accs[d][v];
            }
        }
    }
    __syncthreads();

    if (lane < 16) {
        const int io = ibase + lane;   // == this lane's i (r == lane here)
        // Recompute the diagonal kernel value K(i,i,d) with the IDENTICAL
        // expression used during staging, then subtract it (leave-one-out).
        const float4 tx = reinterpret_cast<const float4*>(trainX)[io];
        float h0 = fmaxf(w1[0]*tx.x + w1[1]*tx.y + w1[2] *tx.z + w1[3] *tx.w, 0.0f);
        float h1 = fmaxf(w1[4]*tx.x + w1[5]*tx.y + w1[6] *tx.z + w1[7] *tx.w, 0.0f);
        float h2 = fmaxf(w1[8]*tx.x + w1[9]*tx.y + w1[10]*tx.z + w1[11]*tx.w, 0.0f);
        const float xs0 = (w2[0]*h0 + w2[1]*h1 + w2[2]*h2) * invh;
        const float xs1 = (w2[3]*h0 + w2[4]*h1 + w2[5]*h2) * invh;
        const float xs2 = (w2[6]*h0 + w2[7]*h1 + w2[8]*h2) * invh;
        const float t0 = xs0 - zs0, t1 = xs1 - zs1, t2 = xs2 - zs2;
        const float kii[3] = { __builtin_amdgcn_exp2f(C2_CONST * t0 * t0),
                               __builtin_amdgcn_exp2f(C2_CONST * t1 * t1),
                               __builtin_amdgcn_exp2f(C2_CONST * t2 * t2) };
#pragma unroll
        for (int d = 0; d < 3; ++d) {
            const float num = sw[d*32 + lane]      - kii[d] * Y[io*3 + d];
            const float den = sw[d*32 + 16 + lane] - kii[d];
            out[io*3 + d] = num / den;
        }
    }
}

extern "C" void kernel_launch(void* const* d_in, const int* in_sizes, int n_in,
                              void* d_out, int out_size, void* d_ws, size_t ws_size,
                              hipStream_t stream) {
    (void)in_sizes; (void)n_in; (void)d_ws; (void)ws_size; (void)out_size;
    const float* x      = (const float*)d_in[0];
    const float* trainX = (const float*)d_in[1];
    const float* Y      = (const float*)d_in[2];
    const float* W1     = (const float*)d_in[3];
    const float* W2     = (const float*)d_in[4];
    const float* h      = (const float*)d_in[5];
    float* out = (float*)d_out;

    nw_loo_kernel<<<NBLOCKS, THREADS, 0, stream>>>(x, trainX, Y, W1, W2, h, out);
}